// HomographEdgeEncoder_72327249264839
// MI455X (gfx1250) — compile-verified
//
#include <hip/hip_runtime.h>
#include <hip/hip_bf16.h>

typedef __attribute__((ext_vector_type(2))) float v2f;
typedef __attribute__((ext_vector_type(8))) float v8f;

#define EMB_DIM 128
#define KCONT   96          // sum of nc over the 9 types (packed unified K)
#define KTOT2   108         // + 9 bias slots, padded to 27 K-steps of 4
#define KSTEPS  27
#define WUSTR   116         // LDS row stride for Wu (even, 52 mod 64 -> 16 banks)
#define ASTR    16          // LDS row stride for attr tile (b128-aligned)

// ---------------- compile-time metadata (DISC/DIMS/CONT are static) --------

struct KTab {
  unsigned char t[KTOT2];    // type owning this unified-K slot (15 = dead pad)
  unsigned char c[KTOT2];    // attr column; 15 = bias slot (ar[15]==1), 14 = zero
  unsigned char off[9];
  unsigned char nc[9];
};
static constexpr KTab genK() {
  KTab K{};
  const int cont[9][12] = {
    {3,6,7,9,10,11,12,13, 0,0,0,0},
    {2,3,4,5,6,7,9,10,11,12,13, 0},
    {2,3,4,5,6,7,9,10,11,12,13, 0},
    {1,4,5,6,7,9,10,11,12,13, 0,0},
    {2,3,4,5,6,7,9,10,11,12,13, 0},
    {1,2,3,4,5,6,7,9,10,11,12,13},
    {2,3,4,5,6,7,9,10,11,12,13, 0},
    {1,2,3,4,5,6,7,9,10,11,12,13},
    {0,1,4,6,7,9,10,11,12,13, 0,0}};
  const int ncs[9] = {8,11,11,10,11,12,11,12,10};
  int kk = 0;
  for (int t = 0; t < 9; ++t) {
    K.off[t] = (unsigned char)kk;
    K.nc[t]  = (unsigned char)ncs[t];
    for (int ci = 0; ci < ncs[t]; ++ci) {
      K.t[kk] = (unsigned char)t;
      K.c[kk] = (unsigned char)cont[t][ci];
      ++kk;
    }
  }
  for (int t = 0; t < 9; ++t) { K.t[kk] = (unsigned char)t; K.c[kk] = 15; ++kk; } // bias
  while (kk < KTOT2) { K.t[kk] = 15; K.c[kk] = 14; ++kk; }                        // dead pad
  return K;
}
static constexpr KTab KM = genK();

// per (type, out-col j): packed {global table id, col-in-table, disc attr col, row width}
struct CTab { unsigned v[9][EMB_DIM]; };
static constexpr CTab genC() {
  CTab M{};
  const int nd[9] = {6,3,3,4,3,2,3,2,4};
  const int dcols[9][6] = {
    {0,1,2,4,5,8},
    {0,1,8,0,0,0},
    {0,1,8,0,0,0},
    {0,2,3,8,0,0},
    {0,1,8,0,0,0},
    {0,8,0,0,0,0},
    {0,1,8,0,0,0},
    {0,8,0,0,0,0},
    {2,3,5,8,0,0}};
  int gbase = 0;
  for (int t = 0; t < 9; ++t) {
    int n = nd[t];
    int per = EMB_DIM / n, rem = EMB_DIM % n;
    int j = 0;
    for (int k = 0; k < n; ++k) {
      int d = per + (k < rem ? 1 : 0);           // _split_dims
      for (int o = 0; o < d; ++o) {
        M.v[t][j++] = (unsigned)(gbase + k) | ((unsigned)o << 5)
                    | ((unsigned)dcols[t][k] << 11) | ((unsigned)d << 15);
      }
    }
    gbase += n;
  }
  return M;
}
static constexpr CTab CM = genC();

// ---------------- kernel ---------------------------------------------------

struct Params {
  const float* edge_attr;
  const float* emb[30];
  const float* lw[9];
  const float* lb[9];
  float* out;
  int nedges;
  int ntiles;
};

__global__ __launch_bounds__(256)
void HomographEdgeEncoder_kernel(Params P) {
  __shared__ float        WuS[EMB_DIM * WUSTR];   // unified weights+bias [j][k']
  __shared__ unsigned     cmapS[9 * EMB_DIM];
  __shared__ const float* embS[30];
  __shared__ float        attrS[8][16 * ASTR];    // per-wave 16x14 attr tile
  __shared__ int          typeS[8][16];

  const int tid = threadIdx.x;
  if (tid < 30) embS[tid] = P.emb[tid];
  for (int flat = tid; flat < EMB_DIM * KTOT2; flat += 256) {
    int j = flat / KTOT2, kk = flat % KTOT2;
    float v = 0.f;
    if (kk < KCONT) {
      int t = KM.t[kk];
      v = P.lw[t][j * KM.nc[t] + (kk - KM.off[t])];
    } else if (kk < KCONT + 9) {
      v = P.lb[kk - KCONT][j];                    // bias folded into GEMM
    }
    WuS[j * WUSTR + kk] = v;
  }
  for (int flat = tid; flat < 9 * EMB_DIM; flat += 256)
    cmapS[flat] = CM.v[flat >> 7][flat & 127];
  __syncthreads();

  const int l    = tid & 31;
  const int w    = tid >> 5;
  const int half = l >> 4;
  const int lm   = l & 15;
  float* aw = attrS[w];
  int*   tw = typeS[w];

  const int nwaves = gridDim.x * 8;
  for (int tile = blockIdx.x * 8 + w; tile < P.ntiles; tile += nwaves) {
    const int e0 = tile * 16;
    const int limit = (P.nedges - e0) * 14;        // tail guard
    // coalesced stage of 16 rows x 14 attrs (224 floats) into LDS
#pragma unroll
    for (int it = 0; it < 7; ++it) {
      int i = it * 32 + l;
      float v = 0.f;
      if (i < limit) v = P.edge_attr[(size_t)e0 * 14 + i];
      aw[(i / 14) * ASTR + (i % 14)] = v;
    }
    // prefetch next tile's attr stream while we compute
    const int ntile = tile + nwaves;
    if (ntile < P.ntiles)
      __builtin_prefetch(P.edge_attr + (size_t)ntile * 16 * 14 + (size_t)l * 16, 0, 1);

    __builtin_amdgcn_wave_barrier();
    asm volatile("" ::: "memory");
    if (l < 16) {
      int t = (int)aw[l * ASTR + 8];
      tw[l] = t < 0 ? 0 : (t > 8 ? 8 : t);
    }
    asm volatile("" ::: "memory");
    __builtin_amdgcn_wave_barrier();

    // lane-resident attr row lm (A-fragment row M = lm); ar[14]=0, ar[15]=1 (bias)
    float ar[16];
    {
      const float4 q0 = *(const float4*)&aw[lm * ASTR + 0];
      const float4 q1 = *(const float4*)&aw[lm * ASTR + 4];
      const float4 q2 = *(const float4*)&aw[lm * ASTR + 8];
      const float2 q3 = *(const float2*)&aw[lm * ASTR + 12];
      ar[0]=q0.x; ar[1]=q0.y; ar[2]=q0.z; ar[3]=q0.w;
      ar[4]=q1.x; ar[5]=q1.y; ar[6]=q1.z; ar[7]=q1.w;
      ar[8]=q2.x; ar[9]=q2.y; ar[10]=q2.z; ar[11]=q2.w;
      ar[12]=q3.x; ar[13]=q3.y; ar[14]=0.f; ar[15]=1.f;
    }
    const int tm = tw[lm];

    v8f C[8];
#pragma unroll
    for (int n = 0; n < 8; ++n) { v8f z = {}; C[n] = z; }

#pragma unroll
    for (int kk = 0; kk < KSTEPS; ++kk) {
      // element v of A/B holds K' = 4*kk + v + 2*half; k-map is compile-time,
      // so both half variants fold to register cndmask chains (no LDS, no branch)
      const int k0 = kk * 4;
      float a0l = (tm == (int)KM.t[k0 + 0]) ? ar[KM.c[k0 + 0]] : 0.f;
      float a1l = (tm == (int)KM.t[k0 + 1]) ? ar[KM.c[k0 + 1]] : 0.f;
      float a0h = (tm == (int)KM.t[k0 + 2]) ? ar[KM.c[k0 + 2]] : 0.f;
      float a1h = (tm == (int)KM.t[k0 + 3]) ? ar[KM.c[k0 + 3]] : 0.f;
      v2f A; A.x = half ? a0h : a0l; A.y = half ? a1h : a1l;
      const int ka = k0 + half * 2;
#pragma unroll
      for (int n = 0; n < 8; ++n) {
        const int j = n * 16 + lm;                 // B col N = lm
        v2f B; B.x = WuS[j * WUSTR + ka]; B.y = WuS[j * WUSTR + ka + 1];
        C[n] = __builtin_amdgcn_wmma_f32_16x16x4_f32(
            false, A, false, B, (short)0, C[n], false, false);
      }
    }

    // epilogue: C fragment (vgpr r, lane l) = (row r + 8*half, col lm)
#pragma unroll
    for (int r = 0; r < 8; ++r) {
      const int m = r + half * 8;
      const int e = e0 + m;
      const bool ok = e < P.nedges;
      const int t = tw[m];
      const unsigned* cmr = &cmapS[t * EMB_DIM];
      float* outp = P.out + (size_t)e * EMB_DIM + lm;
#pragma unroll
      for (int n = 0; n < 8; ++n) {
        const unsigned cm = cmr[n * 16 + lm];
        const int idx = (int)aw[m * ASTR + ((cm >> 11) & 15)];
        const float v = C[n][r]
                      + embS[cm & 31][idx * ((cm >> 15) & 127) + ((cm >> 5) & 63)];
        if (ok) outp[n * 16] = v;
      }
    }
    asm volatile("" ::: "memory");
    __builtin_amdgcn_wave_barrier();   // LDS per-wave in-order: safe WAR vs next tile
  }
}

// ---------------- host glue ------------------------------------------------
// d_in order: [0]=edge_attr, [1..30]=emb tables (30), [31..39]=lin_w, [40..48]=lin_b
extern "C" void kernel_launch(void* const* d_in, const int* in_sizes, int n_in,
                              void* d_out, int out_size, void* d_ws, size_t ws_size,
                              hipStream_t stream) {
  Params P;
  P.edge_attr = (const float*)d_in[0];
  for (int i = 0; i < 30; ++i) P.emb[i] = (const float*)d_in[1 + i];
  for (int i = 0; i < 9;  ++i) P.lw[i]  = (const float*)d_in[31 + i];
  for (int i = 0; i < 9;  ++i) P.lb[i]  = (const float*)d_in[40 + i];
  P.out    = (float*)d_out;
  P.nedges = in_sizes[0] / 14;
  P.ntiles = (P.nedges + 15) / 16;

  int blocks = (P.ntiles + 7) / 8;
  if (blocks > 1024) blocks = 1024;
  hipLaunchKernelGGL(HomographEdgeEncoder_kernel, dim3(blocks), dim3(256), 0, stream, P);
}